// GNNClassifier_88648124990060
// MI455X (gfx1250) — compile-verified
//
#include <hip/hip_runtime.h>

#define N_NODES 100000
#define N_EDGES 3200000
#define EMB 128
#define N_CLASS 10
#define N_GRAPH 1024

typedef float v2f __attribute__((ext_vector_type(2)));
typedef float v8f __attribute__((ext_vector_type(8)));

// ---------------------------------------------------------------------------
// x0[n] = shape_emb[s] + color_emb[c] + pos_emb[p]; 32 lanes x float4 per node
// ---------------------------------------------------------------------------
__global__ void embed_kernel(const int* __restrict__ nf,
                             const float* __restrict__ se,
                             const float* __restrict__ ce,
                             const float* __restrict__ pe,
                             float* __restrict__ x0) {
  int gid = blockIdx.x * blockDim.x + threadIdx.x;
  int n = gid >> 5;
  if (n >= N_NODES) return;
  int c0 = (gid & 31) * 4;
  int s = nf[n * 3 + 0];
  int c = nf[n * 3 + 1];
  int p = nf[n * 3 + 2];
  const float4 a = *(const float4*)(se + s * EMB + c0);
  const float4 b = *(const float4*)(ce + c * EMB + c0);
  const float4 d = *(const float4*)(pe + p * EMB + c0);
  float4 o;
  o.x = a.x + b.x + d.x;
  o.y = a.y + b.y + d.y;
  o.z = a.z + b.z + d.z;
  o.w = a.w + b.w + d.w;
  *(float4*)(x0 + n * EMB + c0) = o;
}

// ---------------------------------------------------------------------------
// deg[dst[e]] += 1  (L2-resident atomics)
// ---------------------------------------------------------------------------
__global__ void degree_kernel(const int* __restrict__ dst, float* __restrict__ deg) {
  int e = blockIdx.x * blockDim.x + threadIdx.x;
  if (e >= N_EDGES) return;
  atomicAdd(deg + dst[e], 1.0f);
}

__global__ void deginv_kernel(float* deg) {
  int i = blockIdx.x * blockDim.x + threadIdx.x;
  if (i >= N_NODES) return;
  float d = deg[i];
  deg[i] = d > 0.0f ? 1.0f / d : 0.0f;  // deg is integral, so 1/max(deg,1)==1/deg
}

// ---------------------------------------------------------------------------
// msg[dst[e]] += x[src[e]] : one wave per edge, lanes cover 128 cols as float4.
// Gather is a coalesced 512B read per edge; adds go to the L2 atomic units.
// ---------------------------------------------------------------------------
__global__ void scatter_kernel(const int* __restrict__ src, const int* __restrict__ dst,
                               const float* __restrict__ x, float* __restrict__ msg) {
  long gid = (long)blockIdx.x * blockDim.x + threadIdx.x;
  int e = (int)(gid >> 5);
  if (e >= N_EDGES) return;
  int c0 = ((int)gid & 31) * 4;
  int s = src[e];
  int d = dst[e];
  const float4 v = *(const float4*)(x + s * EMB + c0);
  float* m = msg + d * EMB + c0;
  atomicAdd(m + 0, v.x);
  atomicAdd(m + 1, v.y);
  atomicAdd(m + 2, v.z);
  atomicAdd(m + 3, v.w);
}

// ---------------------------------------------------------------------------
// out = relu( (msg * deg_inv) @ Wl^T + b + x @ Wr^T )
// One wave owns a 16x16 tile; K=128 via V_WMMA_F32_16X16X4_F32, chaining both
// matmul branches into a single f32 accumulator seeded with the bias.
// A layout (16x4 f32): lane = M (0..15 twice), VGPR0/1 = K+0/K+1, upper half-
//   wave at K+2/K+3 -> contiguous v2f load at x[m*128 + k0 + 2*(lane>>4)].
// B layout (4x16 f32): lane = N, same K split -> contiguous v2f from W[n][k].
// C/D layout: VGPR r holds row M = r + 8*(lane>>4), col N = lane&15.
// ---------------------------------------------------------------------------
__global__ __launch_bounds__(128)
void sage_gemm_kernel(const float* __restrict__ msg, const float* __restrict__ dinv_arr,
                      const float* __restrict__ x,
                      const float* __restrict__ wl, const float* __restrict__ bl,
                      const float* __restrict__ wr, float* __restrict__ out) {
  int wave = threadIdx.x >> 5;
  int lane = threadIdx.x & 31;
  int m0 = (blockIdx.x * 4 + wave) * 16;
  if (m0 >= N_NODES) return;  // wave-uniform: EXEC stays all-ones for WMMA

  int mrow = m0 + (lane & 15);
  int khalf = (lane >> 4) << 1;  // 0 or 2
  float dinv = dinv_arr[mrow];
  const float* xr = x + mrow * EMB + khalf;
  const float* mr = msg + mrow * EMB + khalf;
  int mstore = m0 + ((lane >> 4) << 3);

  for (int nt = 0; nt < 8; ++nt) {
    int ncol = nt * 16 + (lane & 15);
    float bias = bl[ncol];
    v8f acc = {bias, bias, bias, bias, bias, bias, bias, bias};
    const float* wlr = wl + ncol * EMB + khalf;
    const float* wrr = wr + ncol * EMB + khalf;
#pragma unroll
    for (int k0 = 0; k0 < EMB; k0 += 4) {
      v2f aM = *(const v2f*)(mr + k0);
      aM = aM * dinv;                       // fold deg_inv scaling into A
      v2f aX = *(const v2f*)(xr + k0);
      v2f bL = *(const v2f*)(wlr + k0);
      v2f bR = *(const v2f*)(wrr + k0);
      acc = __builtin_amdgcn_wmma_f32_16x16x4_f32(false, aM, false, bL,
                                                  (short)0, acc, false, false);
      acc = __builtin_amdgcn_wmma_f32_16x16x4_f32(false, aX, false, bR,
                                                  (short)0, acc, false, false);
    }
#pragma unroll
    for (int r = 0; r < 8; ++r) {
      float v = acc[r];
      out[(mstore + r) * EMB + ncol] = v > 0.0f ? v : 0.0f;
    }
  }
}

// ---------------------------------------------------------------------------
// pooled[batch[n]] += x[n] ; counts[batch[n]] += 1
// ---------------------------------------------------------------------------
__global__ void pool_kernel(const float* __restrict__ x, const int* __restrict__ batch,
                            float* __restrict__ pooled) {
  int gid = blockIdx.x * blockDim.x + threadIdx.x;
  int n = gid >> 5;
  if (n >= N_NODES) return;
  int c0 = (gid & 31) * 4;
  int g = batch[n];
  const float4 v = *(const float4*)(x + n * EMB + c0);
  float* p = pooled + g * EMB + c0;
  atomicAdd(p + 0, v.x);
  atomicAdd(p + 1, v.y);
  atomicAdd(p + 2, v.z);
  atomicAdd(p + 3, v.w);
}

__global__ void count_kernel(const int* __restrict__ batch, float* __restrict__ counts) {
  int n = blockIdx.x * blockDim.x + threadIdx.x;
  if (n >= N_NODES) return;
  atomicAdd(counts + batch[n], 1.0f);
}

// ---------------------------------------------------------------------------
// out[g, cls] = (pooled[g]/max(cnt,1)) . lin_w[cls] + lin_b[cls]   (1024x10)
// ---------------------------------------------------------------------------
__global__ void head_kernel(const float* __restrict__ pooled, const float* __restrict__ counts,
                            const float* __restrict__ lw, const float* __restrict__ lb,
                            float* __restrict__ out) {
  int t = blockIdx.x * blockDim.x + threadIdx.x;
  if (t >= N_GRAPH * N_CLASS) return;
  int g = t / N_CLASS;
  int cls = t % N_CLASS;
  float cnt = counts[g];
  cnt = cnt > 1.0f ? cnt : 1.0f;
  float inv = 1.0f / cnt;
  float s = 0.0f;
  const float* pr = pooled + g * EMB;
  const float* wr = lw + cls * EMB;
  for (int k = 0; k < EMB; ++k) s += pr[k] * wr[k];
  out[t] = s * inv + lb[cls];
}

// ---------------------------------------------------------------------------
extern "C" void kernel_launch(void* const* d_in, const int* in_sizes, int n_in,
                              void* d_out, int out_size, void* d_ws, size_t ws_size,
                              hipStream_t stream) {
  const int*   node_feats = (const int*)d_in[0];
  const int*   edge_index = (const int*)d_in[1];
  const int*   batch      = (const int*)d_in[2];
  const float* shape_emb  = (const float*)d_in[3];
  const float* color_emb  = (const float*)d_in[4];
  const float* pos_emb    = (const float*)d_in[5];
  const float* w1_l       = (const float*)d_in[6];
  const float* b1_l       = (const float*)d_in[7];
  const float* w1_r       = (const float*)d_in[8];
  const float* w2_l       = (const float*)d_in[9];
  const float* b2_l       = (const float*)d_in[10];
  const float* w2_r       = (const float*)d_in[11];
  const float* lin_w      = (const float*)d_in[12];
  const float* lin_b      = (const float*)d_in[13];

  const int* src = edge_index;            // edge_index[0, :]
  const int* dst = edge_index + N_EDGES;  // edge_index[1, :]

  // Workspace layout (bytes), 256-aligned offsets; x2 reuses the x0 buffer.
  const size_t FEAT_BYTES = (size_t)N_NODES * EMB * sizeof(float);  // 51,200,000
  char* ws = (char*)d_ws;
  float* x0     = (float*)(ws + 0);
  float* msg    = (float*)(ws + FEAT_BYTES);
  float* x1     = (float*)(ws + 2 * FEAT_BYTES);
  float* deg    = (float*)(ws + 3 * FEAT_BYTES);                 // 153,600,000
  float* pooled = (float*)(ws + 3 * FEAT_BYTES + 400128);        // aligned past deg
  float* counts = (float*)(ws + 3 * FEAT_BYTES + 400128 + (size_t)N_GRAPH * EMB * 4);

  hipMemsetAsync(deg, 0, (size_t)N_NODES * sizeof(float), stream);
  hipMemsetAsync(msg, 0, FEAT_BYTES, stream);
  hipMemsetAsync(pooled, 0, (size_t)N_GRAPH * EMB * sizeof(float), stream);
  hipMemsetAsync(counts, 0, (size_t)N_GRAPH * sizeof(float), stream);

  // Embeddings
  embed_kernel<<<(N_NODES * 32 + 255) / 256, 256, 0, stream>>>(
      node_feats, shape_emb, color_emb, pos_emb, x0);

  // Degrees -> inverse degrees (in place)
  degree_kernel<<<(N_EDGES + 255) / 256, 256, 0, stream>>>(dst, deg);
  deginv_kernel<<<(N_NODES + 255) / 256, 256, 0, stream>>>(deg);

  // Layer 1: scatter-add then fused dual-GEMM + bias + ReLU (WMMA f32)
  scatter_kernel<<<(int)(((long)N_EDGES * 32 + 255) / 256), 256, 0, stream>>>(
      src, dst, x0, msg);
  sage_gemm_kernel<<<(N_NODES + 63) / 64, 128, 0, stream>>>(
      msg, deg, x0, w1_l, b1_l, w1_r, x1);

  // Layer 2 (reuse msg buffer, write x2 into the x0 buffer)
  hipMemsetAsync(msg, 0, FEAT_BYTES, stream);
  scatter_kernel<<<(int)(((long)N_EDGES * 32 + 255) / 256), 256, 0, stream>>>(
      src, dst, x1, msg);
  sage_gemm_kernel<<<(N_NODES + 63) / 64, 128, 0, stream>>>(
      msg, deg, x1, w2_l, b2_l, w2_r, x0);

  // Mean pool + linear head
  pool_kernel<<<(N_NODES * 32 + 255) / 256, 256, 0, stream>>>(x0, batch, pooled);
  count_kernel<<<(N_NODES + 255) / 256, 256, 0, stream>>>(batch, counts);
  head_kernel<<<(N_GRAPH * N_CLASS + 127) / 128, 128, 0, stream>>>(
      pooled, counts, lin_w, lin_b, (float*)d_out);
}